// SSMBlock_63720134804186
// MI455X (gfx1250) — compile-verified
//
#include <hip/hip_runtime.h>
#include <hip/hip_bf16.h>
#include <stdint.h>

// Problem constants (from reference setup_inputs)
#define BSZ    2
#define LEN    2048
#define DIM    1024
#define NST    16
// Segmented-scan decomposition
#define NSEG   32            // segments over L
#define SEGLEN 64            // LEN / NSEG
#define TC     16            // timesteps per LDS chunk
#define NCHUNK 4             // SEGLEN / TC
#define DTILE  256           // d-channels per block (== blockDim.x)
#define NDTILE 4             // DIM / DTILE

typedef unsigned int u32x4 __attribute__((ext_vector_type(4)));
typedef int          i32x4 __attribute__((ext_vector_type(4)));
typedef int          i32x8 __attribute__((ext_vector_type(8)));

#if __has_builtin(__builtin_amdgcn_tensor_load_to_lds)
#define HAS_TDM 1
#else
#define HAS_TDM 0
#endif

template <int N>
__device__ __forceinline__ void wait_tensorcnt() {
#if __has_builtin(__builtin_amdgcn_s_wait_tensorcnt)
  __builtin_amdgcn_s_wait_tensorcnt(N);
#else
  asm volatile("s_wait_tensorcnt %0" ::"i"(N) : "memory");
#endif
}

// Issue one TDM 2D tile load: tile_rows x tile_cols f32 elements, row stride
// row_stride_elems, global -> LDS (contiguous, row-major). Wave-uniform args.
__device__ __forceinline__ void tdm_load_2d_f32(uint32_t lds_byte_off,
                                                const float* gptr,
                                                uint32_t tile_cols,
                                                uint32_t tile_rows,
                                                uint32_t row_stride_elems) {
#if HAS_TDM
  uint64_t ga = (uint64_t)(uintptr_t)gptr;
  const uint32_t td0 = 0x7FFFFFFFu;  // tensor_dim0: tile fully in-bounds by construction
  const uint32_t td1 = 0x7FFFFFFFu;  // tensor_dim1
  u32x4 g0;
  g0[0] = 1u;                                        // count=1, user descriptor
  g0[1] = lds_byte_off;                              // lds_addr (bytes)
  g0[2] = (uint32_t)(ga & 0xFFFFFFFFull);            // global_addr[31:0]
  g0[3] = (uint32_t)((ga >> 32) & 0x01FFFFFFull)     // global_addr[56:32]
        | (2u << 30);                                // type = 2 ("image")
  i32x8 g1;
  g1[0] = (int)(2u << 16);                           // data_size=2 -> 4 bytes; mask=0
  g1[1] = (int)((td0 & 0xFFFFu) << 16);              // tensor_dim0[15:0]
  g1[2] = (int)(((td0 >> 16) & 0xFFFFu) | ((td1 & 0xFFFFu) << 16));
  g1[3] = (int)(((td1 >> 16) & 0xFFFFu) | ((tile_cols & 0xFFFFu) << 16)); // tile_dim0
  g1[4] = (int)(tile_rows & 0xFFFFu);                // tile_dim1 (tile_dim2=0)
  g1[5] = (int)row_stride_elems;                     // tensor_dim0_stride[31:0]
  g1[6] = 0;                                         // stride hi / dim1_stride lo
  g1[7] = 0;
  i32x4 z4 = {0, 0, 0, 0};                           // 2D tensor: groups 2/3 unused
  i32x8 z8 = {0, 0, 0, 0, 0, 0, 0, 0};               // unused trailing group (zero-fill)
  __builtin_amdgcn_tensor_load_to_lds(g0, g1, z4, z4, z8, 0);
#else
  (void)lds_byte_off; (void)gptr; (void)tile_cols; (void)tile_rows; (void)row_stride_elems;
#endif
}

// -------- Pass A (FINAL=false): per-segment (P = prod a_bar, Q = H from 0)
// -------- Pass C (FINAL=true):  rerun segment from H0, emit Y
// Double-buffered: TDM for chunk k+1 is issued before computing chunk k, so the
// DMA latency is hidden behind ~TC*NST v_exp_f32 + FMA work per thread.
template <bool FINAL>
__global__ __launch_bounds__(DTILE) void ssm_seg_kernel(
    const float* __restrict__ X, const float* __restrict__ Bm,
    const float* __restrict__ Cm, const float* __restrict__ delta,
    const float* __restrict__ A_log,
    float* __restrict__ P_ws,   // [BSZ,NSEG,DIM,NST]  (pass A writes)
    float* __restrict__ Q_ws,   // pass A: writes Q; pass C: reads H0 (same buffer)
    float* __restrict__ Yout)   // pass C: Y output base
{
  const int tid = threadIdx.x;        // local d index within tile
  const int dt  = blockIdx.x;         // d-tile
  const int seg = blockIdx.y;
  const int b   = blockIdx.z;
  const int d   = dt * DTILE + tid;

  __shared__ __align__(16) float shX [2][TC * DTILE];   // 2 x 16 KB
  __shared__ __align__(16) float shDl[2][TC * DTILE];   // 2 x 16 KB
  __shared__ float shB[2][TC * NST];                    // 2 x 1 KB
  __shared__ float shC[2][TC * NST];                    // 2 x 1 KB

  // Per-thread constants: A row and 1/A (delta cancels in ZOH: B_bar=(a-1)*B/A)
  float Aa[NST], Ainv[NST], H[NST], P[NST];
#pragma unroll
  for (int n = 0; n < NST; ++n) {
    const float a = -__expf(A_log[(size_t)d * NST + n]);
    Aa[n]   = a;
    Ainv[n] = 1.0f / a;
  }
  const size_t chanoff = ((size_t)(b * NSEG + seg) * DIM + d) * NST;
#pragma unroll
  for (int n = 0; n < NST; ++n) {
    H[n] = FINAL ? Q_ws[chanoff + n] : 0.0f;   // pass C starts from segment-entry state
    P[n] = 1.0f;
  }

  const int    t0seg   = seg * SEGLEN;
  const size_t xdbase  = ((size_t)b * LEN + t0seg) * DIM + (size_t)dt * DTILE;
  const size_t bcbase  = ((size_t)b * LEN + t0seg) * NST;

  // Prologue: start chunk 0 into buffer 0
#if HAS_TDM
  if (tid < 32) {
    tdm_load_2d_f32((uint32_t)(uintptr_t)&shX [0][0], X     + xdbase, DTILE, TC, DIM);
    tdm_load_2d_f32((uint32_t)(uintptr_t)&shDl[0][0], delta + xdbase, DTILE, TC, DIM);
  }
#else
  for (int r = 0; r < TC; ++r) {
    shX [0][r * DTILE + tid] = X    [xdbase + (size_t)r * DIM + tid];
    shDl[0][r * DTILE + tid] = delta[xdbase + (size_t)r * DIM + tid];
  }
#endif
  shB[0][tid] = Bm[bcbase + tid];               // [TC,NST] chunk = 256 contiguous floats
  if (FINAL) shC[0][tid] = Cm[bcbase + tid];

  for (int ck = 0; ck < NCHUNK; ++ck) {
    const int cur = ck & 1;
    const int nxt = (ck + 1) & 1;

    __syncthreads();   // all threads done with buffer `nxt` (chunk ck-1) -> safe to refill

    if (ck + 1 < NCHUNK) {   // prefetch chunk ck+1 into the other buffer
      const size_t xdoff = xdbase + (size_t)(ck + 1) * TC * DIM;
#if HAS_TDM
      if (tid < 32) {
        tdm_load_2d_f32((uint32_t)(uintptr_t)&shX [nxt][0], X     + xdoff, DTILE, TC, DIM);
        tdm_load_2d_f32((uint32_t)(uintptr_t)&shDl[nxt][0], delta + xdoff, DTILE, TC, DIM);
      }
#else
      for (int r = 0; r < TC; ++r) {
        shX [nxt][r * DTILE + tid] = X    [xdoff + (size_t)r * DIM + tid];
        shDl[nxt][r * DTILE + tid] = delta[xdoff + (size_t)r * DIM + tid];
      }
#endif
      const size_t bcoff = bcbase + (size_t)(ck + 1) * TC * NST;
      shB[nxt][tid] = Bm[bcoff + tid];
      if (FINAL) shC[nxt][tid] = Cm[bcoff + tid];
    }

#if HAS_TDM
    // Chunk ck's two tile DMAs complete; chunk ck+1's pair may stay in flight.
    if (ck + 1 < NCHUNK) wait_tensorcnt<2>(); else wait_tensorcnt<0>();
#endif
    __syncthreads();   // chunk ck (TDM tiles + B/C rows) visible to all waves

    const int t0 = t0seg + ck * TC;
    for (int t = 0; t < TC; ++t) {
      const float x  = shX [cur][t * DTILE + tid];
      const float dl = shDl[cur][t * DTILE + tid];
      float y = 0.0f;
#pragma unroll
      for (int n = 0; n < NST; ++n) {
        const float ab = __expf(dl * Aa[n]);                      // v_exp_f32
        const float u  = x * (ab - 1.0f) * Ainv[n] * shB[cur][t * NST + n];
        H[n] = __builtin_fmaf(ab, H[n], u);
        if (!FINAL) P[n] *= ab;
        if (FINAL)  y = __builtin_fmaf(shC[cur][t * NST + n], H[n], y);
      }
      if (FINAL) {   // streamed, never re-read: keep it out of near caches
        __builtin_nontemporal_store(y, &Yout[((size_t)b * LEN + (t0 + t)) * DIM + d]);
      }
    }
  }

  if (!FINAL) {
#pragma unroll
    for (int n = 0; n < NST; ++n) {
      P_ws[chanoff + n] = P[n];
      Q_ws[chanoff + n] = H[n];
    }
  }
}

// -------- Pass B: exclusive scan over segments per (b,d,n) channel.
// Reads (P,Q) per segment, overwrites Q with the state entering that segment,
// writes H_final to d_out[0 .. BSZ*DIM*NST).
__global__ __launch_bounds__(256) void ssm_segscan_kernel(
    const float* __restrict__ P_ws, float* __restrict__ Q_ws,
    float* __restrict__ Hfin)
{
  const int idx = blockIdx.x * 256 + threadIdx.x;    // b*DIM*NST + d*NST + n
  if (idx >= BSZ * DIM * NST) return;
  const int b  = idx / (DIM * NST);
  const int dn = idx % (DIM * NST);
  float H = 0.0f;
  for (int s = 0; s < NSEG; ++s) {
    const size_t off = ((size_t)(b * NSEG + s)) * (DIM * NST) + dn;
    const float Ps = P_ws[off];
    const float Qs = Q_ws[off];
    Q_ws[off] = H;                       // segment-entry state for pass C
    H = __builtin_fmaf(Ps, H, Qs);
  }
  Hfin[idx] = H;
}

extern "C" void kernel_launch(void* const* d_in, const int* in_sizes, int n_in,
                              void* d_out, int out_size, void* d_ws, size_t ws_size,
                              hipStream_t stream) {
  (void)in_sizes; (void)n_in; (void)out_size; (void)ws_size;
  const float* X     = (const float*)d_in[0];
  const float* Bm    = (const float*)d_in[1];
  const float* Cm    = (const float*)d_in[2];
  const float* delta = (const float*)d_in[3];
  const float* A_log = (const float*)d_in[4];

  float* out  = (float*)d_out;
  float* Hfin = out;                                   // [BSZ,DIM,NST]
  float* Y    = out + (size_t)BSZ * DIM * NST;         // [BSZ,LEN,DIM]

  float* P_ws = (float*)d_ws;                          // 4 MB
  float* Q_ws = P_ws + (size_t)BSZ * NSEG * DIM * NST; // 4 MB (Q, then H0 in-place)

  dim3 grid(NDTILE, NSEG, BSZ);
  ssm_seg_kernel<false><<<grid, DTILE, 0, stream>>>(X, Bm, Cm, delta, A_log,
                                                    P_ws, Q_ws, nullptr);
  ssm_segscan_kernel<<<(BSZ * DIM * NST + 255) / 256, 256, 0, stream>>>(P_ws, Q_ws, Hfin);
  ssm_seg_kernel<true><<<grid, DTILE, 0, stream>>>(X, Bm, Cm, delta, A_log,
                                                   P_ws, Q_ws, Y);
}